// CNOT_24326694764592
// MI455X (gfx1250) — compile-verified
//
#include <hip/hip_runtime.h>
#include <hip/hip_bf16.h>

// CNOT(13 wires, DIM=2, control=0, target=1) applied to 64 complex64 state
// vectors.  U = P4 (x) I_2048, so Out(4 x 262144 floats) = P4(4x4) @ X.
// The 4x4 block is read live from U (P4[p][q] = Re U[p*2048, q*2048]) and the
// contraction is performed bit-exactly on the matrix unit with
// V_WMMA_F32_16X16X4_F32 (K=4 == contraction dim; entries are 0/1 so the
// f32 MACs are exact).  Traffic: 4 MB in + 4 MB out + 16 U elements -> pure
// HBM-bandwidth kernel (~0.7 us at 23.3 TB/s) vs >=22 us for anything that
// streams the dense 512 MB U.

typedef float v2f __attribute__((ext_vector_type(2)));
typedef float v8f __attribute__((ext_vector_type(8)));

#define DHS     8192            // Hilbert-space dimension
#define GROUPF  (2048 * 64 * 2) // floats per (c,t) group = 262144 (1 MB)
#define NTILES  (GROUPF / 16)   // 16384 column tiles of 16 floats

__global__ __launch_bounds__(256)
void cnot_wmma_kernel(const float* __restrict__ U,
                      const float* __restrict__ X,
                      float* __restrict__ Y) {
    const int lane = threadIdx.x & 31;
    const int mn   = lane & 15;          // M for A/D, N for B/D
    const int k0   = (lane >> 4) ? 2 : 0; // lane-half selects K pair {0,1}/{2,3}

    // ---- A matrix (16x4): rows 0..3 = P4 block of U, rows 4..15 = 0 ----
    // Layout: lanes 0-15 hold {K0,K1} in {VGPR0,VGPR1}; lanes 16-31 hold {K2,K3}.
    v2f a; a.x = 0.0f; a.y = 0.0f;
    if (mn < 4) {
        const size_t rowOff = (size_t)mn * 2048u * (size_t)DHS; // row p*2048 of U
        a.x = U[(rowOff + (size_t)(k0    ) * 2048u) * 2u];      // Re U[p*2048, q*2048]
        a.y = U[(rowOff + (size_t)(k0 + 1) * 2048u) * 2u];
    }

    const int wavesPerBlock = blockDim.x >> 5;
    const int gw = blockIdx.x * wavesPerBlock + (threadIdx.x >> 5);
    const int nw = gridDim.x * wavesPerBlock;

    for (int t = gw; t < NTILES; t += nw) {
        const int col = t * 16 + mn;

        // ---- B matrix (4x16): lane = N, lane-half selects K pair ----
        v2f b;
        b.x = X[(size_t)(k0    ) * GROUPF + (size_t)col];
        b.y = X[(size_t)(k0 + 1) * GROUPF + (size_t)col];

        v8f c = {0.f, 0.f, 0.f, 0.f, 0.f, 0.f, 0.f, 0.f};
        // D = A x B + 0  (EXEC is all-1s here: uniform control flow)
        v8f d = __builtin_amdgcn_wmma_f32_16x16x4_f32(
            /*neg_a=*/false, a, /*neg_b=*/false, b,
            /*c_mod=*/(short)0, c, /*reuse_a=*/false, /*reuse_b=*/false);

        // D rows 0..3 live in d[0..3] on lanes 0-15 (N = lane).
        if (lane < 16) {
            Y[(size_t)0 * GROUPF + (size_t)col] = d[0];
            Y[(size_t)1 * GROUPF + (size_t)col] = d[1];
            Y[(size_t)2 * GROUPF + (size_t)col] = d[2];
            Y[(size_t)3 * GROUPF + (size_t)col] = d[3];
        }
    }
}

extern "C" void kernel_launch(void* const* d_in, const int* in_sizes, int n_in,
                              void* d_out, int out_size, void* d_ws, size_t ws_size,
                              hipStream_t stream) {
    (void)in_sizes; (void)n_in; (void)out_size; (void)d_ws; (void)ws_size;

    const float* U = (const float*)d_in[0]; // complex64 viewed as interleaved f32
    const float* X = (const float*)d_in[1];
    float*       Y = (float*)d_out;

    // 512 blocks x 8 waves = 4096 waves; each wave handles 4 of 16384 tiles.
    cnot_wmma_kernel<<<512, 256, 0, stream>>>(U, X, Y);
}